// IGFE_SpatialAttention_6511170421404
// MI455X (gfx1250) — compile-verified
//
#include <hip/hip_runtime.h>
#include <math.h>

typedef __attribute__((ext_vector_type(2))) float v2f;
typedef __attribute__((ext_vector_type(8))) float v8f;

#define BATCH    8
#define CH       128
#define HH       64
#define WWID     64
#define HEADS    4
#define HEAD_DIM 32
#define HPOOL    32
#define NTOK     1024
#define TOPK     32
#define NEG_INF  (-3.0e38f)

__device__ __forceinline__ int iclamp(int v, int lo, int hi) {
    return v < lo ? lo : (v > hi ? hi : v);
}

// ---------------------------------------------------------------------------
// 1) 2x2 average pool: (B,C,64,64) -> (B,C,32,32)
// ---------------------------------------------------------------------------
__global__ void pool_k(const float* __restrict__ in, float* __restrict__ out, int total) {
    int i = blockIdx.x * blockDim.x + threadIdx.x;
    if (i >= total) return;
    int x  = i & 31;
    int y  = (i >> 5) & 31;
    int bc = i >> 10;
    const float* p = in + (size_t)bc * 4096 + (2 * y) * 64 + 2 * x;
    out[i] = 0.25f * (p[0] + p[1] + p[64] + p[65]);
}

// ---------------------------------------------------------------------------
// 2) Generic fp32 WMMA GEMM (1x1 conv): Y[b,o,p] = bias[o] + sum_i W[o,i]*X[b,i,p]
//    One 16x16 output tile per wave, K stepped by 4 via v_wmma_f32_16x16x4_f32.
//    A(16x4) layout: lane<16 -> M=lane, vgpr0 = K=k0+0 / (lanes 16-31) K=k0+2.
//    C/D layout: vgpr r -> M = r + 8*(lane>>4), N = lane&15.
// ---------------------------------------------------------------------------
__global__ void gemm16_wmma(const float* __restrict__ Wm, const float* __restrict__ Xm,
                            const float* __restrict__ bias, float* __restrict__ Ym,
                            int M, int K, int Npos) {
    int lane = threadIdx.x & 31;
    int wave = threadIdx.x >> 5;
    int tile = blockIdx.x * (blockDim.x >> 5) + wave;
    int ntN  = Npos >> 4;
    int mTile = tile / ntN;
    int nTile = tile - mTile * ntN;
    if (mTile >= (M >> 4)) return;
    int b = blockIdx.z;
    const float* X = Xm + (size_t)b * K * Npos;
    float*       Y = Ym + (size_t)b * M * Npos;

    int mBase = mTile << 4, nBase = nTile << 4;
    int half = lane >> 4;       // 0 or 1
    int l16  = lane & 15;
    int m = mBase + l16;
    int n = nBase + l16;

    v8f c = {};
    for (int k0 = 0; k0 < K; k0 += 4) {
        int kk = k0 + 2 * half;
        v2f a, bm;
        a.x  = Wm[(size_t)m * K + kk];
        a.y  = Wm[(size_t)m * K + kk + 1];
        bm.x = X[(size_t)kk * Npos + n];
        bm.y = X[(size_t)(kk + 1) * Npos + n];
        c = __builtin_amdgcn_wmma_f32_16x16x4_f32(false, a, false, bm,
                                                  (short)0, c, false, false);
    }
#pragma unroll
    for (int r = 0; r < 8; ++r) {
        int mr = mBase + r + 8 * half;
        Y[(size_t)mr * Npos + n] = c[r] + bias[mr];
    }
}

// ---------------------------------------------------------------------------
// 3) Depthwise 3x3 on q1 (B,C,32,32) + repack to head layout (B,H,N,d)
// ---------------------------------------------------------------------------
__global__ void q2_dw_pack(const float* __restrict__ q1, const float* __restrict__ w,
                           const float* __restrict__ bias, float* __restrict__ qh, int total) {
    int i = blockIdx.x * blockDim.x + threadIdx.x;
    if (i >= total) return;
    int x = i & 31, y = (i >> 5) & 31;
    int c = (i >> 10) & (CH - 1);
    int b = i >> 17;
    const float* in = q1 + ((size_t)b * CH + c) * NTOK;
    const float* wc = w + c * 9;
    float acc = bias[c];
#pragma unroll
    for (int dy = -1; dy <= 1; ++dy)
#pragma unroll
        for (int dx = -1; dx <= 1; ++dx) {
            int yy = y + dy, xx = x + dx;
            if (yy >= 0 && yy < 32 && xx >= 0 && xx < 32)
                acc += in[yy * 32 + xx] * wc[(dy + 1) * 3 + (dx + 1)];
        }
    int h = c >> 5, d = c & 31, n = y * 32 + x;
    qh[(((size_t)b * HEADS + h) * NTOK + n) * HEAD_DIM + d] = acc;
}

// ---------------------------------------------------------------------------
// 4) Grouped 3x3 (2-in/2-out per group) on kv1 (B,256,32,32), split+pack K and V
// ---------------------------------------------------------------------------
__global__ void kv2_gconv_pack(const float* __restrict__ kv1, const float* __restrict__ w,
                               const float* __restrict__ bias, float* __restrict__ kh,
                               float* __restrict__ vh, int total) {
    int i = blockIdx.x * blockDim.x + threadIdx.x;
    if (i >= total) return;
    int x  = i & 31, y = (i >> 5) & 31;
    int oc = (i >> 10) & 255;
    int b  = i >> 18;
    int g  = oc >> 1;
    float acc = bias[oc];
#pragma unroll
    for (int ic = 0; ic < 2; ++ic) {
        const float* in = kv1 + ((size_t)b * 256 + 2 * g + ic) * NTOK;
        const float* wc = w + ((size_t)oc * 2 + ic) * 9;
#pragma unroll
        for (int dy = -1; dy <= 1; ++dy)
#pragma unroll
            for (int dx = -1; dx <= 1; ++dx) {
                int yy = y + dy, xx = x + dx;
                if (yy >= 0 && yy < 32 && xx >= 0 && xx < 32)
                    acc += in[yy * 32 + xx] * wc[(dy + 1) * 3 + (dx + 1)];
            }
    }
    int n = y * 32 + x;
    if (oc < CH) {
        int c = oc;
        kh[(((size_t)b * HEADS + (c >> 5)) * NTOK + n) * HEAD_DIM + (c & 31)] = acc;
    } else {
        int c = oc - CH;
        vh[(((size_t)b * HEADS + (c >> 5)) * NTOK + n) * HEAD_DIM + (c & 31)] = acc;
    }
}

// ---------------------------------------------------------------------------
// 5) Depthwise 5x5 lepe on x (B,C,64,64), pad 2
// ---------------------------------------------------------------------------
__global__ void lepe_dw5(const float* __restrict__ x, const float* __restrict__ w,
                         const float* __restrict__ bias, float* __restrict__ out, int total) {
    int i = blockIdx.x * blockDim.x + threadIdx.x;
    if (i >= total) return;
    int xx = i & 63, yy = (i >> 6) & 63;
    int c = (i >> 12) & (CH - 1);
    int b = i >> 19;
    const float* in = x + ((size_t)b * CH + c) * 4096;
    const float* wc = w + c * 25;
    float acc = bias[c];
#pragma unroll
    for (int dy = -2; dy <= 2; ++dy)
#pragma unroll
        for (int dx = -2; dx <= 2; ++dx) {
            int y2 = yy + dy, x2 = xx + dx;
            if (y2 >= 0 && y2 < 64 && x2 >= 0 && x2 < 64)
                acc += in[y2 * 64 + x2] * wc[(dy + 2) * 5 + (dx + 2)];
        }
    out[i] = acc;
}

// ---------------------------------------------------------------------------
// 6) Attention: 4-wave workgroup per (b,h,16-query tile).
//    Phase 1: each wave computes a disjoint quarter of the 64 key tiles with
//             fp32 WMMA; logits + rel_pos land in a shared 64KB LDS buffer.
//             (64KB/WG -> 5 WGs = 20 waves resident per WGP for the HBM-bound
//              rel_pos stream.)
//    Phase 2: each wave runs top-32 / softmax / V-gather on 4 disjoint rows.
//             In-wave LDS ordering via volatile (DS ops are in-order per wave),
//             so no per-selection barrier is needed.
// ---------------------------------------------------------------------------
__global__ void __launch_bounds__(128) attn_topk_wmma(
    const float* __restrict__ qh, const float* __restrict__ kh,
    const float* __restrict__ vh, const float* __restrict__ rel,
    const float* __restrict__ temp, float* __restrict__ attnout) {
    __shared__ float sc[16][NTOK];   // 64 KB of the WGP's 320 KB LDS

    int lane  = threadIdx.x & 31;
    int wave  = threadIdx.x >> 5;    // 0..3
    int blk   = blockIdx.x;          // bh * 64 + qTile
    int qTile = blk & 63;
    int bh    = blk >> 6;
    int h     = bh & (HEADS - 1);
    int qBase = qTile << 4;
    int half  = lane >> 4, l16 = lane & 15;

    const float* Qb = qh + (size_t)bh * NTOK * HEAD_DIM;
    const float* Kb = kh + (size_t)bh * NTOK * HEAD_DIM;
    const float* Vb = vh + (size_t)bh * NTOK * HEAD_DIM;
    const float* Rb = rel + (size_t)bh * NTOK * NTOK;
    float tscale = temp[h];

    // Preload the full 16x32 Q tile as WMMA-A fragments (8 K-steps of 4)
    v2f a[8];
    {
        const float* qrow = Qb + (size_t)(qBase + l16) * HEAD_DIM;
#pragma unroll
        for (int t = 0; t < 8; ++t) {
            int kk = 4 * t + 2 * half;
            a[t].x = qrow[kk];
            a[t].y = qrow[kk + 1];
        }
    }

    // Phase 1: wave w covers key tiles [w*16, w*16+16)
    for (int t16 = 0; t16 < 16; ++t16) {
        int kt      = wave * 16 + t16;
        int keyBase = kt << 4;
        const float* krow = Kb + (size_t)(keyBase + l16) * HEAD_DIM;
        v8f c = {};
#pragma unroll
        for (int t = 0; t < 8; ++t) {
            int kk = 4 * t + 2 * half;
            v2f bm;
            bm.x = krow[kk];
            bm.y = krow[kk + 1];
            c = __builtin_amdgcn_wmma_f32_16x16x4_f32(false, a[t], false, bm,
                                                      (short)0, c, false, false);
        }
        // Hint the streaming rel_pos read for the next tile (global_prefetch_b8)
        if (t16 + 1 < 16)
            __builtin_prefetch(&Rb[(size_t)(qBase + half * 8) * NTOK + keyBase + 16], 0, 0);
#pragma unroll
        for (int r = 0; r < 8; ++r) {
            int m   = r + 8 * half;
            int key = keyBase + l16;
            sc[m][key] = tscale * c[r] + Rb[(size_t)(qBase + m) * NTOK + key];
        }
    }
    __syncthreads();

    // Phase 2: wave w owns rows [w*4, w*4+4)
    for (int qq = 0; qq < 4; ++qq) {
        int q = wave * 4 + qq;
        volatile float* row = &sc[q][0];   // in-wave ordering, no register caching
        float myV = 0.f;
        int   myI = 0;
        for (int s = 0; s < TOPK; ++s) {
            float lv = NEG_INF;
            int   li = 0x7fffffff;
#pragma unroll 4
            for (int t = 0; t < 32; ++t) {
                int j = (t << 5) + lane;       // conflict-free LDS access
                float v = row[j];
                if (v > lv) { lv = v; li = j; }
            }
#pragma unroll
            for (int off = 16; off > 0; off >>= 1) {
                float ov = __shfl_xor(lv, off, 32);
                int   oi = __shfl_xor(li, off, 32);
                if (ov > lv || (ov == lv && oi < li)) { lv = ov; li = oi; }
            }
            if (lane == s) { myV = lv; myI = li; }   // lane s keeps s-th largest
            if (lane == 0) row[li] = NEG_INF;        // exclude from next pass
        }
        // softmax over the 32 selected logits (lane = selection rank)
        float mx  = __shfl(myV, 0, 32);
        float e   = __expf(myV - mx);
        float sum = e;
#pragma unroll
        for (int off = 16; off > 0; off >>= 1) sum += __shfl_xor(sum, off, 32);
        float wgt = e / sum;

        // gather V rows; lane = output dim d
        float acc = 0.f;
        for (int s = 0; s < TOPK; ++s) {
            float wv  = __shfl(wgt, s, 32);
            int   idx = __shfl(myI, s, 32);
            acc += wv * Vb[(size_t)idx * HEAD_DIM + lane];
        }
        attnout[((size_t)bh * NTOK + qBase + q) * HEAD_DIM + lane] = acc;
    }
}

// ---------------------------------------------------------------------------
// 7) from_heads + bilinear 2x upsample (half-pixel, edge clamp) + lepe add
// ---------------------------------------------------------------------------
__global__ void upsample_add(const float* __restrict__ attnout, const float* __restrict__ lepe,
                             float* __restrict__ pre, int total) {
    int i = blockIdx.x * blockDim.x + threadIdx.x;
    if (i >= total) return;
    int x = i & 63, y = (i >> 6) & 63;
    int c = (i >> 12) & (CH - 1);
    int b = i >> 19;

    float sy = 0.5f * y - 0.25f, sx = 0.5f * x - 0.25f;
    float fy = floorf(sy), fx = floorf(sx);
    float wy = sy - fy, wx = sx - fx;
    int y0 = iclamp((int)fy, 0, 31), y1 = iclamp((int)fy + 1, 0, 31);
    int x0 = iclamp((int)fx, 0, 31), x1 = iclamp((int)fx + 1, 0, 31);

    int h = c >> 5, d = c & 31;
    const float* S = attnout + ((size_t)(b * HEADS + h) * NTOK) * HEAD_DIM + d;
    float v00 = S[(size_t)(y0 * 32 + x0) * HEAD_DIM];
    float v01 = S[(size_t)(y0 * 32 + x1) * HEAD_DIM];
    float v10 = S[(size_t)(y1 * 32 + x0) * HEAD_DIM];
    float v11 = S[(size_t)(y1 * 32 + x1) * HEAD_DIM];
    float v = (1.f - wy) * ((1.f - wx) * v00 + wx * v01)
            + wy        * ((1.f - wx) * v10 + wx * v11);
    pre[i] = v + lepe[i];
}

// ---------------------------------------------------------------------------
// launch
// ---------------------------------------------------------------------------
extern "C" void kernel_launch(void* const* d_in, const int* in_sizes, int n_in,
                              void* d_out, int out_size, void* d_ws, size_t ws_size,
                              hipStream_t stream) {
    const float* x      = (const float*)d_in[0];
    const float* y      = (const float*)d_in[1];
    const float* rel    = (const float*)d_in[2];
    const float* q1_w   = (const float*)d_in[3];
    const float* q1_b   = (const float*)d_in[4];
    const float* q2_w   = (const float*)d_in[5];
    const float* q2_b   = (const float*)d_in[6];
    const float* kv1_w  = (const float*)d_in[7];
    const float* kv1_b  = (const float*)d_in[8];
    const float* kv2_w  = (const float*)d_in[9];
    const float* kv2_b  = (const float*)d_in[10];
    const float* lepe_w = (const float*)d_in[11];
    const float* lepe_b = (const float*)d_in[12];
    const float* out_w  = (const float*)d_in[13];
    const float* out_b  = (const float*)d_in[14];
    const float* temp   = (const float*)d_in[15];
    float* out = (float*)d_out;

    float* ws = (float*)d_ws;
    size_t off = 0;
    float* xp      = ws + off; off += (size_t)BATCH * CH * NTOK;       // 4 MB
    float* yp      = ws + off; off += (size_t)BATCH * CH * NTOK;       // 4 MB
    float* q1buf   = ws + off; off += (size_t)BATCH * CH * NTOK;       // 4 MB
    float* kv1buf  = ws + off; off += (size_t)BATCH * 2 * CH * NTOK;   // 8 MB
    float* qh      = ws + off; off += (size_t)BATCH * CH * NTOK;       // 4 MB
    float* kh      = ws + off; off += (size_t)BATCH * CH * NTOK;       // 4 MB
    float* vh      = ws + off; off += (size_t)BATCH * CH * NTOK;       // 4 MB
    float* attnout = ws + off; off += (size_t)BATCH * CH * NTOK;       // 4 MB
    float* lepe    = ws + off; off += (size_t)BATCH * CH * 4096;       // 16 MB
    float* pre     = ws + off; off += (size_t)BATCH * CH * 4096;       // 16 MB
    (void)ws_size; (void)in_sizes; (void)n_in; (void)out_size;

    int npool = BATCH * CH * NTOK;
    pool_k<<<(npool + 255) / 256, 256, 0, stream>>>(x, xp, npool);
    pool_k<<<(npool + 255) / 256, 256, 0, stream>>>(y, yp, npool);

    // q1: 128x128 GEMM over 1024 positions per batch (512 tiles -> 128 blocks of 4 waves)
    dim3 gq((CH / 16) * (NTOK / 16) / 4, 1, BATCH);
    gemm16_wmma<<<gq, 128, 0, stream>>>(q1_w, xp, q1_b, q1buf, CH, CH, NTOK);

    // kv1: 256x128 GEMM
    dim3 gkv((2 * CH / 16) * (NTOK / 16) / 4, 1, BATCH);
    gemm16_wmma<<<gkv, 128, 0, stream>>>(kv1_w, yp, kv1_b, kv1buf, 2 * CH, CH, NTOK);

    q2_dw_pack<<<(npool + 255) / 256, 256, 0, stream>>>(q1buf, q2_w, q2_b, qh, npool);

    int nkv2 = BATCH * 2 * CH * NTOK;
    kv2_gconv_pack<<<(nkv2 + 255) / 256, 256, 0, stream>>>(kv1buf, kv2_w, kv2_b, kh, vh, nkv2);

    int nfull = BATCH * CH * 4096;
    lepe_dw5<<<(nfull + 255) / 256, 256, 0, stream>>>(x, lepe_w, lepe_b, lepe, nfull);

    // attention: 4-wave workgroup per (b,h,16-query tile)
    attn_topk_wmma<<<BATCH * HEADS * (NTOK / 16), 128, 0, stream>>>(qh, kh, vh, rel, temp, attnout);

    upsample_add<<<(nfull + 255) / 256, 256, 0, stream>>>(attnout, lepe, pre, nfull);

    // final 1x1 conv: 128x128 GEMM over 4096 positions per batch
    dim3 go((CH / 16) * (4096 / 16) / 4, 1, BATCH);
    gemm16_wmma<<<go, 128, 0, stream>>>(out_w, pre, out_b, out, CH, CH, 4096);
}